// DDFPack_48885317763784
// MI455X (gfx1250) — compile-verified
//
#include <hip/hip_runtime.h>
#include <math.h>

// Problem constants (match reference)
#define BB   8
#define CC   256
#define HH   56
#define WW   56
#define HWX  (HH*WW)       // 3136
#define TAPS 9             // K*K
#define MIDD 51            // int(256*0.2)
#define CKK  (CC*TAPS)     // 2304
#define GAIN_OVER_K (1.41421356237309515f / 3.0f)

typedef __attribute__((ext_vector_type(2))) float v2f;
typedef __attribute__((ext_vector_type(8))) float v8f;

// ---------------------------------------------------------------------------
// K1: pooled[b*C+c] = mean_{h,w} x[b,c,h,w].  One block per (b,c) plane.
// ---------------------------------------------------------------------------
__global__ void k_pool(const float* __restrict__ x, float* __restrict__ pooled) {
    __shared__ float red[256];
    const int bc = blockIdx.x;                     // 0 .. B*C-1
    const float* p = x + (size_t)bc * HWX;
    float s = 0.f;
    for (int i = threadIdx.x; i < HWX; i += 256) s += p[i];
    red[threadIdx.x] = s;
    __syncthreads();
    for (int off = 128; off > 0; off >>= 1) {
        if (threadIdx.x < off) red[threadIdx.x] += red[threadIdx.x + off];
        __syncthreads();
    }
    if (threadIdx.x == 0) pooled[bc] = red[0] * (1.0f / (float)HWX);
}

// ---------------------------------------------------------------------------
// K2: tiny MLP (256 -> 51 relu -> 2304), per-channel tap-normalize, * fn_std
//     One block (256 threads) per batch; thread t == channel c in phase 3.
// ---------------------------------------------------------------------------
__global__ void k_cf(const float* __restrict__ pooled,
                     const float* __restrict__ w1, const float* __restrict__ b1,
                     const float* __restrict__ w2, const float* __restrict__ b2,
                     const float* __restrict__ fn_std,
                     float* __restrict__ cf) {
    __shared__ float ps[CC];
    __shared__ float y1[64];
    const int b = blockIdx.x;
    const int t = threadIdx.x;
    ps[t] = pooled[b * CC + t];
    __syncthreads();
    if (t < MIDD) {
        float a = b1[t];
        const float* wr = w1 + t * CC;
        for (int c = 0; c < CC; ++c) a += ps[c] * wr[c];
        y1[t] = a > 0.f ? a : 0.f;
    }
    __syncthreads();
    // thread t is channel c: compute its 9 taps, normalize (ddof=1), scale
    float v[TAPS];
    float mean = 0.f;
    for (int k = 0; k < TAPS; ++k) {
        const int n = t * TAPS + k;
        float a = b2[n];
        const float* wr = w2 + n * MIDD;
        for (int m = 0; m < MIDD; ++m) a += y1[m] * wr[m];
        v[k] = a;
        mean += a;
    }
    mean *= (1.f / TAPS);
    float var = 0.f;
    for (int k = 0; k < TAPS; ++k) { float d = v[k] - mean; var += d * d; }
    const float stdv = sqrtf(var * (1.f / (TAPS - 1)));
    const float inv = 1.f / (stdv + 1e-10f);
    for (int k = 0; k < TAPS; ++k) {
        const int n = t * TAPS + k;
        cf[(size_t)b * CKK + n] = (v[k] - mean) * inv * fn_std[n];
    }
}

// ---------------------------------------------------------------------------
// K3: sf_raw[b, n, p] = sum_c ws[n,c] * x[b,c,p] + bs[n]
//     GEMM via V_WMMA_F32_16X16X4_F32: A = ws padded to 16 rows (LDS),
//     B = 4-channel x 16-pixel tile of x.  One wave per 16-pixel tile,
//     8 waves per block.  64 wmma issues per wave (K = 256 / 4).
// ---------------------------------------------------------------------------
#define WSLD 257   // LDS row stride (floats) to spread banks
__global__ void k_sf(const float* __restrict__ x, const float* __restrict__ ws,
                     const float* __restrict__ bs, float* __restrict__ sf_raw) {
    __shared__ float wsp[16 * WSLD];
    // Stage A (ws) into LDS, zero-padded to 16 rows.
    for (int idx = threadIdx.x; idx < 16 * CC; idx += 256) {
        const int r = idx >> 8, k = idx & 255;
        wsp[r * WSLD + k] = (r < TAPS) ? ws[r * CC + k] : 0.f;
    }
    __syncthreads();

    const int wave = threadIdx.x >> 5;
    const int lane = threadIdx.x & 31;
    const int half = lane >> 4;       // 0: lanes 0-15, 1: lanes 16-31
    const int ln   = lane & 15;
    const int tile = blockIdx.x * 8 + wave;      // 0 .. B*(3136/16)-1
    const int b    = tile / (HWX / 16);
    const int p0   = (tile % (HWX / 16)) * 16;
    const float* xb = x + (size_t)b * CC * HWX;

    v8f acc = {};
    for (int k0 = 0; k0 < CC; k0 += 4) {
        const int ka = k0 + 2 * half;            // this half-wave's K pair
        v2f a, bm;
        // A fragment: VGPR0 = K even, VGPR1 = K odd; halves cover K pairs.
        a.x = wsp[ln * WSLD + ka];
        a.y = wsp[ln * WSLD + ka + 1];
        // B fragment: B[k][n] = x[b, k, p0+n]  (pixels along lanes)
        bm.x = xb[(size_t)ka       * HWX + p0 + ln];
        bm.y = xb[(size_t)(ka + 1) * HWX + p0 + ln];
        acc = __builtin_amdgcn_wmma_f32_16x16x4_f32(
            /*neg_a=*/false, a, /*neg_b=*/false, bm,
            /*c_mod=*/(short)0, acc, /*reuse_a=*/false, /*reuse_b=*/false);
    }
    // D layout: VGPR r, lanes 0-15 -> M=r, lanes 16-31 -> M=r+8; N = ln.
    #pragma unroll
    for (int r = 0; r < 8; ++r) {
        const int M = r + 8 * half;
        if (M < TAPS) {
            sf_raw[((size_t)b * TAPS + M) * HWX + p0 + ln] = acc[r] + bs[M];
        }
    }
}

// ---------------------------------------------------------------------------
// K4: normalize sf over the 9 taps per (b, pixel), ddof=1, * (sqrt(2)/3).
//     In place; each thread owns its 9 values.
// ---------------------------------------------------------------------------
__global__ void k_sfnorm(float* __restrict__ sf) {
    const int idx = blockIdx.x * 256 + threadIdx.x;   // b*HWX + p
    if (idx >= BB * HWX) return;
    const int b = idx / HWX, p = idx % HWX;
    float* base = sf + (size_t)b * TAPS * HWX + p;
    float v[TAPS];
    float mean = 0.f;
    #pragma unroll
    for (int t = 0; t < TAPS; ++t) { v[t] = base[(size_t)t * HWX]; mean += v[t]; }
    mean *= (1.f / TAPS);
    float var = 0.f;
    #pragma unroll
    for (int t = 0; t < TAPS; ++t) { float d = v[t] - mean; var += d * d; }
    const float stdv = sqrtf(var * (1.f / (TAPS - 1)));
    const float sc = GAIN_OVER_K / (stdv + 1e-10f);
    #pragma unroll
    for (int t = 0; t < TAPS; ++t) base[(size_t)t * HWX] = (v[t] - mean) * sc;
}

// ---------------------------------------------------------------------------
// K5: DDF combine.  out[b,c,i,j] = sum_taps x[b,c,i+u-1,j+v-1]*cf*sf.
//     One thread per output element; x/sf/cf are L2-resident by now.
// ---------------------------------------------------------------------------
__global__ void k_ddf(const float* __restrict__ x, const float* __restrict__ cf,
                      const float* __restrict__ sf, float* __restrict__ out) {
    const int o = blockIdx.x * 256 + threadIdx.x;     // exact grid
    const int j = o % WW;
    const int i = (o / WW) % HH;
    const int c = (o / HWX) % CC;
    const int b = o / (HWX * CC);
    const float* sfb = sf + (size_t)b * TAPS * HWX + i * WW + j;
    const float* cfp = cf + ((size_t)b * CC + c) * TAPS;
    const float* xb  = x  + ((size_t)b * CC + c) * HWX;
    float acc = 0.f;
    #pragma unroll
    for (int u = 0; u < 3; ++u) {
        const int ii = i + u - 1;
        if (ii >= 0 && ii < HH) {
            #pragma unroll
            for (int v = 0; v < 3; ++v) {
                const int jj = j + v - 1;
                if (jj >= 0 && jj < WW) {
                    const int t = u * 3 + v;
                    acc += xb[ii * WW + jj] * cfp[t] * sfb[(size_t)t * HWX];
                }
            }
        }
    }
    out[o] = acc;
}

// ---------------------------------------------------------------------------
extern "C" void kernel_launch(void* const* d_in, const int* in_sizes, int n_in,
                              void* d_out, int out_size, void* d_ws, size_t ws_size,
                              hipStream_t stream) {
    const float* x      = (const float*)d_in[0];
    const float* w1     = (const float*)d_in[1];
    const float* b1     = (const float*)d_in[2];
    const float* w2     = (const float*)d_in[3];
    const float* b2     = (const float*)d_in[4];
    const float* ws     = (const float*)d_in[5];
    const float* bs     = (const float*)d_in[6];
    const float* fn_std = (const float*)d_in[7];
    float* out = (float*)d_out;

    // Workspace layout (floats): pooled[2048] | cf[18432] | sf[225792]
    float* wsf    = (float*)d_ws;
    float* pooled = wsf;
    float* cf     = wsf + (BB * CC);
    float* sf     = wsf + (BB * CC) + (BB * CKK);

    k_pool  <<<BB * CC,            256, 0, stream>>>(x, pooled);
    k_cf    <<<BB,                 256, 0, stream>>>(pooled, w1, b1, w2, b2, fn_std, cf);
    k_sf    <<<(BB * HWX / 16) / 8,256, 0, stream>>>(x, ws, bs, sf);      // 196 blocks
    k_sfnorm<<<(BB * HWX) / 256,   256, 0, stream>>>(sf);                 // 98 blocks
    k_ddf   <<<(BB * CC * HWX)/256,256, 0, stream>>>(x, cf, sf, out);     // 25088 blocks
}